// LieParameterization_66640712564954
// MI455X (gfx1250) — compile-verified
//
#include <hip/hip_runtime.h>
#include <stdint.h>

// LieParameterization: out[b] = expm(sum_s params[b,s]*SON_BASES[s]), 8x8 skew-symmetric.
// Memory-bound (193 MB @ 23.3 TB/s ~ 8.3 us floor) -> minimize per-wave instruction issues.
// f32 WMMA scaling-and-squaring: Xs = A/2^6, Taylor-8 Horner (8 matmuls), 6 squarings.
//
// Per wave: FOUR batch elements = two packed problems. Each packed problem:
//   A = [X0;X1] (16x8), B = [T0|T1] (8x16), D = A*B -> useful diag blocks X0T0/X1T1.
//   2x V_WMMA_F32_16X16X4_F32 per 16x16x8 matmul; 28 matmuls -> 56 WMMA per wave.
//
// WMMA f32 register layouts (CDNA5 ISA 7.12.2, wave32):
//  A 16x4:  lanes 0-15: M=lane, vgpr0=K0, vgpr1=K1 ; lanes 16-31: M=lane-16, vgpr0=K2, vgpr1=K3
//  B 4x16:  vgpr0: lanes0-15 K=0 / lanes16-31 K=2 ; vgpr1: K=1 / K=3  (N = lane&15)
//  C/D 16x16: vgpr r: lanes0-15 = D[r][lane], lanes16-31 = D[r+8][lane-16]
//  => useful D entries U[r][n] live in vgpr r at lanes {0-7 (n=0..7), 24-31 (n=8..15)}.
//
// LDS tile is COLUMN-major, U[col][row], column stride 12 floats (48B = 16B-multiple):
//  - extraction store: one lane owns a whole column -> 2x ds_store_b128 per tile
//  - B loads: rows {2h,2h+1} contiguous within a column -> 2x ds_load_b64 per tile
// No explicit LDS fences: same-wave DS ops are in-order in hardware (ISA 7.3); the
// backend inserts the register-consumption waits, and may-alias analysis on lane-variant
// LDS addresses blocks reordering.

typedef __attribute__((ext_vector_type(2))) float v2f;
typedef __attribute__((ext_vector_type(4))) float v4f;
typedef __attribute__((ext_vector_type(8))) float v8f;

#define NUM_BASES 28
#define NSQ       6             // squarings (scale 2^-6)
#define NTAYLOR   8             // Horner iterations k=8..1 (Taylor order 8)
#define CS        12            // column stride (floats): 48B, keeps b128/b64 aligned
#define TILE_F    (32 * CS)     // 16 data columns + 16 dummy columns (branchless stores)
#define WAVES_PER_BLOCK 8

__device__ __forceinline__ int son_index(int i, int j) {  // basis index for (i,j), i<j
  return i * 8 - (i * (i + 1)) / 2 + (j - i - 1);
}

__global__ void __launch_bounds__(32 * WAVES_PER_BLOCK)
lie_expm_kernel(const float* __restrict__ params, float* __restrict__ out,
                int nwaves, int batch) {
  __shared__ float tiles[WAVES_PER_BLOCK][2 * TILE_F];

  const int lane = threadIdx.x & 31;
  const int wid  = threadIdx.x >> 5;
  const int w    = blockIdx.x * WAVES_PER_BLOCK + wid;  // wave index: elements 4w..4w+3
  if (w >= nwaves) return;                              // wave-uniform: EXEC stays full

  float* __restrict__ U0 = &tiles[wid][0];
  uint32_t t1off = TILE_F;                 // opaque tile-1 offset: forces its own DS base
  asm volatile("" : "+v"(t1off));          // register, keeping immediate offsets small
  float* __restrict__ U1 = U0 + t1off;

  const int h  = lane >> 4;   // lane half (K sub-pair select in A/B layouts)
  const int m  = lane & 15;   // A-row / B-column
  const int im = m & 7;       // row within 8x8 block
  const int tb = m >> 3;      // which matrix of the packed pair

  // ---- stage 112 contiguous params (4 elements) into LDS via async copy -------------
  {
    const int fbase = w * 112 + lane * 4;               // float index of this lane's 16B chunk
    if (lane < 28 && fbase + 4 <= batch * NUM_BASES) {
      const uint32_t lds_addr = (uint32_t)(uintptr_t)(U0 + lane * 4);  // low 32b = LDS offset
      const uint32_t goff     = (uint32_t)fbase * 4u;                  // byte offset vs base
      asm volatile("global_load_async_to_lds_b128 %0, %1, %2 offset:0"
                   :: "v"(lds_addr), "v"(goff), "s"(params) : "memory");
    }
  }
  asm volatile("s_wait_asynccnt 0x0" ::: "memory");  // async->LDS visible before DS reads

  // ---- build Xs = A/64 in A-layout for both packed problems (branchless sign/scatter)
  v2f xa[2][2];  // [problem][K-panel]
  #pragma unroll
  for (int t = 0; t < 2; ++t) {
    const int eb = t * 2 + tb;  // element within wave: 0..3
    #pragma unroll
    for (int p = 0; p < 2; ++p) {
      #pragma unroll
      for (int v = 0; v < 2; ++v) {
        const int i = im, j = p * 4 + 2 * h + v;
        int sidx = (i < j) ? son_index(i, j) : son_index(j, i);
        if (sidx < 0) sidx = 0;  // diagonal lanes: clamped, multiplied by 0
        const float sgn = (i < j) ? (1.0f / 64.0f) : ((i > j) ? (-1.0f / 64.0f) : 0.0f);
        xa[t][p][v] = U0[eb * NUM_BASES + sidx] * sgn;
      }
    }
  }

  // ---- per-lane extraction constants ------------------------------------------------
  const bool act  = (lane < 8) || (lane >= 24);
  const int  ncol = (lane < 8) ? lane : (lane - 16);        // owned U column (active lanes)
  const int  offc = (act ? ncol : (16 + (lane - 8))) * CS;  // inactive -> dummy columns
  v4f dlo, dhi;                                             // +I pattern on both diag blocks
  #pragma unroll
  for (int r = 0; r < 4; ++r) {
    dlo[r] = (ncol == r     || ncol == r + 8)  ? 1.0f : 0.0f;
    dhi[r] = (ncol == r + 4 || ncol == r + 12) ? 1.0f : 0.0f;
  }

  // ---- init U = I: each lane stores its column as two b128 --------------------------
  *(v4f*)(U0 + offc) = dlo;  *(v4f*)(U0 + offc + 4) = dhi;
  *(v4f*)(U1 + offc) = dlo;  *(v4f*)(U1 + offc + 4) = dhi;

  const int bB = m * CS + 2 * h;  // B-load base: col m, rows {2h,2h+1} (+4 for panel 1)

  // ---- Horner: T <- I + (Xs*T)/k for k=8..1 (k=1 yields E = exp(Xs) approx) ---------
  #pragma unroll
  for (int k = NTAYLOR; k >= 1; --k) {
    const float invk = 1.0f / (float)k;
    #pragma unroll
    for (int t = 0; t < 2; ++t) {
      float* __restrict__ U = t ? U1 : U0;
      const v2f b0 = *(const v2f*)(U + bB);      // rows {2h,2h+1} of col m
      const v2f b1 = *(const v2f*)(U + bB + 4);  // rows {4+2h,5+2h}
      v8f d = {};
      d = __builtin_amdgcn_wmma_f32_16x16x4_f32(false, xa[t][0], false, b0, (short)0, d, false, false);
      d = __builtin_amdgcn_wmma_f32_16x16x4_f32(false, xa[t][1], false, b1, (short)0, d, false, false);
      v4f lo, hi;
      #pragma unroll
      for (int r = 0; r < 4; ++r) {
        lo[r] = __builtin_fmaf(d[r],     invk, dlo[r]);
        hi[r] = __builtin_fmaf(d[r + 4], invk, dhi[r]);
      }
      *(v4f*)(U + offc) = lo;  *(v4f*)(U + offc + 4) = hi;  // 2x ds_store_b128
    }
  }

  // ---- 6 squarings: Y <- Y*Y --------------------------------------------------------
  const int c0 = 2 * h + 8 * tb;  // A-panel-0 columns {c0, c0+1} of [Y0;Y1] row im
  #pragma unroll
  for (int s = 0; s < NSQ; ++s) {
    #pragma unroll
    for (int t = 0; t < 2; ++t) {
      float* __restrict__ U = t ? U1 : U0;
      v2f a0, a1;
      a0[0] = U[c0 * CS + im];        a0[1] = U[(c0 + 1) * CS + im];  // 2addr pairs
      a1[0] = U[(c0 + 4) * CS + im];  a1[1] = U[(c0 + 5) * CS + im];
      const v2f b0 = *(const v2f*)(U + bB);
      const v2f b1 = *(const v2f*)(U + bB + 4);
      v8f d = {};
      d = __builtin_amdgcn_wmma_f32_16x16x4_f32(false, a0, false, b0, (short)0, d, false, false);
      d = __builtin_amdgcn_wmma_f32_16x16x4_f32(false, a1, false, b1, (short)0, d, false, false);
      const v4f lo = { d[0], d[1], d[2], d[3] };
      const v4f hi = { d[4], d[5], d[6], d[7] };
      *(v4f*)(U + offc) = lo;  *(v4f*)(U + offc + 4) = hi;  // 2x ds_store_b128
    }
  }

  // ---- write out 4 contiguous 8x8 results: 2 floats/lane -> coalesced 256B b64 ------
  float* __restrict__ outp = out + (size_t)w * 256;
  #pragma unroll
  for (int q = 0; q < 4; ++q) {                // q = batch element within wave (uniform)
    const int f  = q * 64 + lane * 2;          // 2 consecutive output floats per lane
    const int rr = (f >> 3) & 7;               // = lane>>2
    const int cc = (f & 7) + 8 * (q & 1);      // U column (+8 for second matrix of pair)
    const float* __restrict__ U = (q < 2) ? U0 : U1;
    v2f val;
    val[0] = U[cc * CS + rr];                  // 2addr pair (stride 12 dwords)
    val[1] = U[(cc + 1) * CS + rr];
    if (w * 4 + q < batch)
      *(v2f*)(outp + f) = val;
  }
}

extern "C" void kernel_launch(void* const* d_in, const int* in_sizes, int n_in,
                              void* d_out, int out_size, void* d_ws, size_t ws_size,
                              hipStream_t stream) {
  const float* params = (const float*)d_in[0];
  float* out = (float*)d_out;
  const int batch  = in_sizes[0] / NUM_BASES;
  const int nwaves = (batch + 3) / 4;
  const int blocks = (nwaves + WAVES_PER_BLOCK - 1) / WAVES_PER_BLOCK;
  lie_expm_kernel<<<blocks, 32 * WAVES_PER_BLOCK, 0, stream>>>(params, out, nwaves, batch);
}